// ICA_59785944760581
// MI455X (gfx1250) — compile-verified
//
#include <hip/hip_runtime.h>
#include <hip/hip_bf16.h>
#include <math.h>

// ---------------------------------------------------------------------------
// ICA (JADE) pipeline for gfx1250 / MI455X.
//   x (B,3,T,H,W) f32 -> xm (B,T,3), Wica (B,3,3), S (B,3,T)
// Strategy: single bandwidth-bound streaming pass over x (b128 vector loads),
// then tiny on-device JADE stages. Batched Gram matrices (Rx, R, Qm) use
// V_WMMA_F32_16X16X4_F32 (one wave per batch, K-loop over time).
// ---------------------------------------------------------------------------

typedef __attribute__((ext_vector_type(2))) float v2f;
typedef __attribute__((ext_vector_type(8))) float v8f;

#define LAM2 10000.0f  // LAM^2, LAM = 100

// ------------------------- kernel 1: spatial mean ---------------------------
// Bandwidth-critical pass: 128-bit coalesced loads when rows are 16B-aligned.
__global__ void mean_kernel(const float* __restrict__ x, float* __restrict__ xm_out,
                            float* __restrict__ xmBT, int B, int T, int HW, float invHW) {
    const int idx = blockIdx.x;            // idx = (b*3 + c)*T + t
    const int t  = idx % T;
    const int bc = idx / T;
    const int c  = bc % 3;
    const int b  = bc / 3;
    const float* src = x + (size_t)idx * HW;
    float sum = 0.f;
    if ((HW & 3) == 0) {
        // each row starts at a multiple of HW floats -> 16B aligned
        const float4* src4 = (const float4*)src;
        const int nv = HW >> 2;
        for (int i = threadIdx.x; i < nv; i += blockDim.x) {
            float4 v = src4[i];                       // global_load_b128
            sum += (v.x + v.y) + (v.z + v.w);
        }
    } else {
        for (int i = threadIdx.x; i < HW; i += blockDim.x) sum += src[i];
    }
    __shared__ float sd[256];
    sd[threadIdx.x] = sum;
    __syncthreads();
    for (int o = 128; o > 0; o >>= 1) {
        if (threadIdx.x < o) sd[threadIdx.x] += sd[threadIdx.x + o];
        __syncthreads();
    }
    if (threadIdx.x == 0) {
        float m = sd[0] * invHW;
        xm_out[((size_t)b * T + t) * 3 + c] = m;   // xm (B,T,3)
        xmBT[(size_t)idx] = m;                     // [b][c][t] work layout
    }
}

// --------------- kernel 2: banded LDL^T of (I + lam^2 D2^T D2) -------------
__global__ void penta_factor(float* __restrict__ l1, float* __restrict__ l2,
                             float* __restrict__ dd, int T) {
    if (threadIdx.x != 0 || blockIdx.x != 0) return;
    const float dco[3] = {1.f, -2.f, 1.f};
    for (int i = 0; i < T; ++i) {
        float a0 = 0.f;                                  // M[i][i]
        for (int r = i - 2; r <= i; ++r) {
            if (r < 0 || r > T - 3) continue;
            float v = dco[i - r];
            a0 += v * v;
        }
        a0 = 1.f + LAM2 * a0;
        float A1m = 0.f;                                 // M[i-1][i]
        if (i >= 1) {
            for (int r = i - 3; r <= i - 1; ++r) {
                if (r < 0 || r > T - 3) continue;
                int c1 = i - 1 - r, c2 = i - r;
                if (c1 < 0 || c1 > 2 || c2 < 0 || c2 > 2) continue;
                A1m += dco[c1] * dco[c2];
            }
            A1m *= LAM2;
        }
        float A2m = 0.f;                                 // M[i-2][i]
        if (i >= 2) {
            int r = i - 2;
            if (r >= 0 && r <= T - 3) A2m = LAM2;        // dco[0]*dco[2] = 1
        }
        float L2 = (i >= 2) ? A2m / dd[i - 2] : 0.f;
        float L1 = 0.f;
        if (i >= 1) {
            float num = A1m - ((i >= 2) ? L2 * l1[i - 1] * dd[i - 2] : 0.f);
            L1 = num / dd[i - 1];
        }
        float D = a0 - ((i >= 1) ? L1 * L1 * dd[i - 1] : 0.f)
                     - ((i >= 2) ? L2 * L2 * dd[i - 2] : 0.f);
        l1[i] = L1;
        l2[i] = L2;
        dd[i] = (fabsf(D) > 1e-20f) ? D : 1e-20f;
    }
}

// --------- kernel 3: detrend det = xm - M^{-1} xm (penta solve) -------------
__global__ void penta_solve(const float* __restrict__ xmBT, const float* __restrict__ l1,
                            const float* __restrict__ l2, const float* __restrict__ dd,
                            float* __restrict__ detB, int BC, int T) {
    int idx = blockIdx.x * blockDim.x + threadIdx.x;
    if (idx >= BC) return;
    const float* rhs = xmBT + (size_t)idx * T;
    float* y = detB + (size_t)idx * T;
    float zm1 = 0.f, zm2 = 0.f;                          // forward: L z = rhs
    for (int i = 0; i < T; ++i) {
        float z = rhs[i] - l1[i] * zm1 - l2[i] * zm2;
        y[i] = z; zm2 = zm1; zm1 = z;
    }
    float yp1 = 0.f, yp2 = 0.f;                          // back: L^T y = z/d
    for (int i = T - 1; i >= 0; --i) {
        float w = y[i] / dd[i];
        float yi = w - ((i + 1 < T) ? l1[i + 1] * yp1 : 0.f)
                     - ((i + 2 < T) ? l2[i + 2] * yp2 : 0.f);
        y[i] = yi; yp2 = yp1; yp1 = yi;
    }
    for (int i = 0; i < T; ++i) y[i] = rhs[i] - y[i];    // det = xm - y
}

// ------------------- kernel 4: per-(b,c) mean / std (ddof=1) ----------------
__global__ void stats_kernel(const float* __restrict__ detB, float* __restrict__ stats, int T) {
    int bc = blockIdx.x;
    const float* d = detB + (size_t)bc * T;
    __shared__ float sbuf[256];
    float s = 0.f;
    for (int i = threadIdx.x; i < T; i += blockDim.x) s += d[i];
    sbuf[threadIdx.x] = s; __syncthreads();
    for (int o = 128; o > 0; o >>= 1) {
        if (threadIdx.x < o) sbuf[threadIdx.x] += sbuf[threadIdx.x + o];
        __syncthreads();
    }
    float mu = sbuf[0] / (float)T;
    __syncthreads();
    float v = 0.f;
    for (int i = threadIdx.x; i < T; i += blockDim.x) { float e = d[i] - mu; v += e * e; }
    sbuf[threadIdx.x] = v; __syncthreads();
    for (int o = 128; o > 0; o >>= 1) {
        if (threadIdx.x < o) sbuf[threadIdx.x] += sbuf[threadIdx.x + o];
        __syncthreads();
    }
    if (threadIdx.x == 0) {
        int den = (T > 1) ? (T - 1) : 1;
        stats[bc * 2]     = mu;
        stats[bc * 2 + 1] = sqrtf(sbuf[0] / (float)den);
    }
}

// ------------- kernel 5: standardize into zero-padded 16 x Tp rows ----------
__global__ void normalize_pad(const float* __restrict__ detB, const float* __restrict__ stats,
                              float* __restrict__ Xpad, int T, int Tp) {
    int bc = blockIdx.x;
    int b = bc / 3, c = bc % 3;
    float mu = stats[bc * 2];
    float sd = stats[bc * 2 + 1];
    float inv = 1.f / ((sd > 1e-20f) ? sd : 1e-20f);
    const float* d = detB + (size_t)bc * T;
    float* dst = Xpad + ((size_t)b * 16 + c) * Tp;
    for (int t = threadIdx.x; t < T; t += blockDim.x) dst[t] = (d[t] - mu) * inv;
}

// ---------------- kernel 6: WMMA Gram  G = Xp * Xp^T * invT -----------------
// One wave per batch. Xp is 16 x Tp (zero-padded). F32 WMMA 16x16x4, K over Tp.
__global__ void __launch_bounds__(32) gram16_wmma(const float* __restrict__ Xp,
                                                  float* __restrict__ G,
                                                  int Tp, float invT) {
    const int b    = blockIdx.x;
    const int lane = threadIdx.x;      // 0..31, full wave
    const int half = lane >> 4;        // 0/1
    const int row  = lane & 15;        // M for A, N for B/C
    const float* Xb = Xp + (size_t)b * 16 * Tp;
#if __has_builtin(__builtin_amdgcn_wmma_f32_16x16x4_f32)
    v8f acc = {};
    for (int k = 0; k < Tp; k += 4) {
        // A 16x4 f32: VGPR0 = K = 2*half, VGPR1 = K = 2*half+1 (lane rows 0..15)
        // B 4x16  = X^T  => B fragment equals A fragment for a Gram product.
        v2f a;
        a.x = Xb[row * Tp + k + 2 * half];
        a.y = Xb[row * Tp + k + 2 * half + 1];
        acc = __builtin_amdgcn_wmma_f32_16x16x4_f32(false, a, false, a,
                                                    (short)0, acc, false, false);
    }
    float* Gb = G + (size_t)b * 256;
    #pragma unroll
    for (int r = 0; r < 8; ++r)                 // C/D: VGPR r -> M = r + 8*half
        Gb[(r + 8 * half) * 16 + row] = acc[r] * invT;
#else
    // fallback (should not be taken on gfx1250)
    float accs[8] = {0};
    for (int k = 0; k < Tp; ++k)
        for (int r = 0; r < 8; ++r)
            accs[r] += Xb[(r + 8 * half) * Tp + k] * Xb[row * Tp + k];
    float* Gb = G + (size_t)b * 256;
    for (int r = 0; r < 8; ++r) Gb[(r + 8 * half) * 16 + row] = accs[r] * invT;
#endif
}

// ------------------------ small dense eigh (Jacobi) -------------------------
__device__ void jacobi_eigh(float* A, float* V, int n, int sweeps) {
    for (int i = 0; i < n; ++i)
        for (int j = 0; j < n; ++j) V[i * n + j] = (i == j) ? 1.f : 0.f;
    for (int s = 0; s < sweeps; ++s) {
        for (int p = 0; p < n - 1; ++p) {
            for (int q = p + 1; q < n; ++q) {
                float apq = A[p * n + q];
                if (fabsf(apq) < 1e-30f) continue;
                float app = A[p * n + p], aqq = A[q * n + q];
                float tau = (aqq - app) / (2.f * apq);
                float t = ((tau >= 0.f) ? 1.f : -1.f) / (fabsf(tau) + sqrtf(1.f + tau * tau));
                float c = 1.f / sqrtf(1.f + t * t);
                float sn = t * c;
                for (int k = 0; k < n; ++k) {
                    float akp = A[k * n + p], akq = A[k * n + q];
                    A[k * n + p] = c * akp - sn * akq;
                    A[k * n + q] = sn * akp + c * akq;
                }
                for (int k = 0; k < n; ++k) {
                    float apk = A[p * n + k], aqk = A[q * n + k];
                    A[p * n + k] = c * apk - sn * aqk;
                    A[q * n + k] = sn * apk + c * aqk;
                }
                for (int k = 0; k < n; ++k) {
                    float vkp = V[k * n + p], vkq = V[k * n + q];
                    V[k * n + p] = c * vkp - sn * vkq;
                    V[k * n + q] = sn * vkp + c * vkq;
                }
            }
        }
    }
}

// ------------- kernel 7: whitening W = U d^{-1/2} U^T, IW = U d^{1/2} U^T ---
__global__ void whiten3(const float* __restrict__ Rx, float* __restrict__ Wm,
                        float* __restrict__ IWm, int B) {
    int b = blockIdx.x * blockDim.x + threadIdx.x;
    if (b >= B) return;
    float A[9], U[9];
    for (int i = 0; i < 3; ++i)
        for (int j = 0; j < 3; ++j) A[i * 3 + j] = Rx[(size_t)b * 256 + i * 16 + j];
    jacobi_eigh(A, U, 3, 12);
    float d[3] = {fmaxf(A[0], 1e-12f), fmaxf(A[4], 1e-12f), fmaxf(A[8], 1e-12f)};
    for (int i = 0; i < 3; ++i) {
        for (int j = 0; j < 3; ++j) {
            float w = 0.f, iw = 0.f;
            for (int k = 0; k < 3; ++k) {
                float sq = sqrtf(d[k]);
                w  += U[i * 3 + k] * (1.f / sq) * U[j * 3 + k];
                iw += U[i * 3 + k] * sq         * U[j * 3 + k];
            }
            Wm[(size_t)b * 9 + i * 3 + j]  = w;
            IWm[(size_t)b * 9 + i * 3 + j] = iw;
        }
    }
}

// ------------------- kernel 8: Y = W * Xn (into padded rows) ----------------
__global__ void ycompute(const float* __restrict__ Xpad, const float* __restrict__ Wm,
                         float* __restrict__ Ypad, int B, int T, int Tp) {
    int i = blockIdx.x * blockDim.x + threadIdx.x;
    if (i >= B * T) return;
    int b = i / T, t = i % T;
    const float* X = Xpad + (size_t)b * 16 * Tp;
    const float* W = Wm + (size_t)b * 9;
    float x0 = X[t], x1 = X[Tp + t], x2 = X[2 * Tp + t];
    float* Y = Ypad + (size_t)b * 16 * Tp;
    Y[t]          = W[0] * x0 + W[1] * x1 + W[2] * x2;
    Y[Tp + t]     = W[3] * x0 + W[4] * x1 + W[5] * x2;
    Y[2 * Tp + t] = W[6] * x0 + W[7] * x1 + W[8] * x2;
}

// -------------- kernel 9: cumulant rows P[l*3+k][t] = Y_l Y_k ---------------
__global__ void pcompute(const float* __restrict__ Ypad, float* __restrict__ Ppad,
                         int B, int T, int Tp) {
    int i = blockIdx.x * blockDim.x + threadIdx.x;
    if (i >= B * T) return;
    int b = i / T, t = i % T;
    const float* Y = Ypad + (size_t)b * 16 * Tp;
    float y[3] = {Y[t], Y[Tp + t], Y[2 * Tp + t]};
    float* P = Ppad + (size_t)b * 16 * Tp;
    for (int l = 0; l < 3; ++l)
        for (int k = 0; k < 3; ++k)
            P[(size_t)(l * 3 + k) * Tp + t] = y[l] * y[k];
}

// --------- kernel 10: JADE core (9x9 eigh, 50 Jacobi sweeps, Wica) ----------
__global__ void jade_core(const float* __restrict__ Qraw, const float* __restrict__ Rm,
                          const float* __restrict__ IWm, float* __restrict__ Vm,
                          float* __restrict__ outW, int B, int T) {
    int b = blockIdx.x * blockDim.x + threadIdx.x;
    if (b >= B) return;
    const float seuil = 1.f / (sqrtf((float)T) * 100.f);
    float R[9];
    for (int i = 0; i < 3; ++i)
        for (int j = 0; j < 3; ++j) R[i * 3 + j] = Rm[(size_t)b * 256 + i * 16 + j];
    float Qm[81];
    for (int row = 0; row < 9; ++row) {
        int l = row / 3, k = row % 3;
        for (int col = 0; col < 9; ++col) {
            int jj = col / 3, ii = col % 3;
            Qm[row * 9 + col] = Qraw[(size_t)b * 256 + row * 16 + col]
                - R[ii * 3 + jj] * R[l * 3 + k]
                - R[ii * 3 + k]  * R[l * 3 + jj]
                - R[ii * 3 + l]  * R[jj * 3 + k];
        }
    }
    float Uq[81];
    jacobi_eigh(Qm, Uq, 9, 20);
    float lam[9];
    for (int i = 0; i < 9; ++i) lam[i] = Qm[i * 9 + i];
    bool used[9] = {false, false, false, false, false, false, false, false, false};
    float M[27];                                         // 3 x 9
    for (int blk = 0; blk < 3; ++blk) {                  // top-3 by |lambda|
        int best = 0; float bv = -1.f;
        for (int j = 0; j < 9; ++j)
            if (!used[j] && fabsf(lam[j]) > bv) { bv = fabsf(lam[j]); best = j; }
        used[best] = true;
        float la = lam[best];
        for (int i = 0; i < 3; ++i)
            for (int j = 0; j < 3; ++j)
                M[i * 9 + blk * 3 + j] = la * Uq[(i * 3 + j) * 9 + best];
    }
    float V[9] = {1, 0, 0, 0, 1, 0, 0, 0, 1};
    bool run = true;
    for (int sweep = 0; sweep < 50; ++sweep) {
        bool enc = false;
        for (int p = 0; p < 2; ++p) {
            for (int q = p + 1; q < 3; ++q) {
                float G00 = 0, G01 = 0, G02 = 0, G11 = 0, G12 = 0, G22 = 0;
                for (int n = 0; n < 3; ++n) {
                    int Ip = p + 3 * n, Iq = q + 3 * n;
                    float g0 = M[p * 9 + Ip] - M[q * 9 + Iq];
                    float g1 = M[p * 9 + Iq];
                    float g2 = M[q * 9 + Ip];
                    G00 += g0 * g0; G01 += g0 * g1; G02 += g0 * g2;
                    G11 += g1 * g1; G12 += g1 * g2; G22 += g2 * g2;
                }
                // temp = Re(Bc G Bc^H) block-diagonalizes for real G:
                float t00 = G00, t01 = G01 + G02;
                float t11 = G11 + G22 + 2.f * G12;
                float t22 = G11 + G22 - 2.f * G12;
                float mid = (t00 + t11) * 0.5f;
                float rad = sqrtf((t00 - t11) * (t00 - t11) * 0.25f + t01 * t01);
                float lmax = mid + rad;
                float ax, bx;
                if (lmax >= t22) {
                    float vx = t01, vy = lmax - t00;
                    float nn = sqrtf(vx * vx + vy * vy);
                    if (nn < 1e-20f) { vx = lmax - t11; vy = t01; nn = sqrtf(vx * vx + vy * vy); }
                    if (nn < 1e-20f) { vx = 1.f; vy = 0.f; nn = 1.f; }
                    ax = vx / nn; bx = vy / nn;
                    if (ax < 0.f) { ax = -ax; bx = -bx; }
                } else { ax = 0.f; bx = 0.f; }
                float r  = fmaxf(sqrtf(ax * ax + bx * bx), 1e-12f);
                float c  = ax / r, s = bx / r;
                bool apply = run && (fabsf(s) > seuil);
                enc = enc || apply;
                float c_ = apply ? c : 1.f;
                float s_ = apply ? s : 0.f;
                for (int i = 0; i < 3; ++i) {            // V columns p,q
                    float vp = V[i * 3 + p], vq = V[i * 3 + q];
                    V[i * 3 + p] =  c_ * vp + s_ * vq;
                    V[i * 3 + q] = -s_ * vp + c_ * vq;
                }
                for (int j = 0; j < 9; ++j) {            // M rows p,q
                    float mp = M[p * 9 + j], mq = M[q * 9 + j];
                    M[p * 9 + j] =  c_ * mp + s_ * mq;
                    M[q * 9 + j] = -s_ * mp + c_ * mq;
                }
                for (int n = 0; n < 3; ++n) {            // M cols Ip,Iq
                    int Ip = p + 3 * n, Iq = q + 3 * n;
                    for (int i = 0; i < 3; ++i) {
                        float a0 = M[i * 9 + Ip], b0 = M[i * 9 + Iq];
                        M[i * 9 + Ip] =  c_ * a0 + s_ * b0;
                        M[i * 9 + Iq] = -s_ * a0 + c_ * b0;
                    }
                }
            }
        }
        run = enc;
    }
    const float* IW = IWm + (size_t)b * 9;               // A = IW * V
    float A[9];
    for (int i = 0; i < 3; ++i)
        for (int j = 0; j < 3; ++j) {
            float ssum = 0.f;
            for (int k = 0; k < 3; ++k) ssum += IW[i * 3 + k] * V[k * 3 + j];
            A[i * 3 + j] = ssum;
        }
    float det = A[0] * (A[4] * A[8] - A[5] * A[7])
              - A[1] * (A[3] * A[8] - A[5] * A[6])
              + A[2] * (A[3] * A[7] - A[4] * A[6]);
    float id = 1.f / ((fabsf(det) > 1e-20f) ? det : ((det >= 0.f) ? 1e-20f : -1e-20f));
    float Wi[9];
    Wi[0] =  (A[4] * A[8] - A[5] * A[7]) * id;
    Wi[1] = -(A[1] * A[8] - A[2] * A[7]) * id;
    Wi[2] =  (A[1] * A[5] - A[2] * A[4]) * id;
    Wi[3] = -(A[3] * A[8] - A[5] * A[6]) * id;
    Wi[4] =  (A[0] * A[8] - A[2] * A[6]) * id;
    Wi[5] = -(A[0] * A[5] - A[2] * A[3]) * id;
    Wi[6] =  (A[3] * A[7] - A[4] * A[6]) * id;
    Wi[7] = -(A[0] * A[7] - A[1] * A[6]) * id;
    Wi[8] =  (A[0] * A[4] - A[1] * A[3]) * id;
    for (int i = 0; i < 9; ++i) {
        outW[(size_t)b * 9 + i] = Wi[i];
        Vm[(size_t)b * 9 + i]   = V[i];
    }
}

// ------------------------- kernel 11: S = V^T Y -----------------------------
__global__ void s_kernel(const float* __restrict__ Ypad, const float* __restrict__ Vm,
                         float* __restrict__ outS, int B, int T, int Tp) {
    int i = blockIdx.x * blockDim.x + threadIdx.x;
    if (i >= B * T) return;
    int b = i / T, t = i % T;
    const float* Y = Ypad + (size_t)b * 16 * Tp;
    const float* V = Vm + (size_t)b * 9;
    float y0 = Y[t], y1 = Y[Tp + t], y2 = Y[2 * Tp + t];
    outS[((size_t)b * 3 + 0) * T + t] = V[0] * y0 + V[3] * y1 + V[6] * y2;
    outS[((size_t)b * 3 + 1) * T + t] = V[1] * y0 + V[4] * y1 + V[7] * y2;
    outS[((size_t)b * 3 + 2) * T + t] = V[2] * y0 + V[5] * y1 + V[8] * y2;
}

// ---------------------------------------------------------------------------
extern "C" void kernel_launch(void* const* d_in, const int* in_sizes, int n_in,
                              void* d_out, int out_size, void* d_ws, size_t ws_size,
                              hipStream_t stream) {
    const float* x = (const float*)d_in[0];
    float* out = (float*)d_out;
    float* ws  = (float*)d_ws;
    const long in_n = (long)in_sizes[0];

    // Infer (B, T, HW):  out_size = B*(6T+9),  in_n = B*3*T*HW
    int B = 1, T = 4; long HW = 1; bool found = false;
    for (int b = 1; b <= 8192 && !found; ++b) {
        if (out_size % b) continue;
        long per = (long)out_size / b;
        if (per <= 9 || (per - 9) % 6) continue;
        long t = (per - 9) / 6;
        if (t < 4) continue;
        if (in_n % ((long)b * 3 * t)) continue;
        B = b; T = (int)t; HW = in_n / ((long)b * 3 * t); found = true;
    }
    if (!found) {
        B = 1;
        T = (out_size > 9) ? (int)(((long)out_size - 9) / 6) : 4;
        if (T < 4) T = 4;
        HW = in_n / ((long)3 * T);
        if (HW < 1) HW = 1;
    }
    const int Tp = (T + 3) & ~3;

    // Workspace layout (floats)
    size_t o = 0;
    const size_t xmBT  = o; o += (size_t)B * 3 * T;
    const size_t detB  = o; o += (size_t)B * 3 * T;
    const size_t facL1 = o; o += (size_t)T;
    const size_t facL2 = o; o += (size_t)T;
    const size_t facDD = o; o += (size_t)T;
    const size_t stats = o; o += (size_t)B * 6;
    const size_t Xpad  = o; o += (size_t)B * 16 * Tp;
    const size_t Ypad  = o; o += (size_t)B * 16 * Tp;
    const size_t Ppad  = o; o += (size_t)B * 16 * Tp;
    const size_t RxO   = o; o += (size_t)B * 256;
    const size_t RmO   = o; o += (size_t)B * 256;
    const size_t QrO   = o; o += (size_t)B * 256;
    const size_t WmO   = o; o += (size_t)B * 9;
    const size_t IWO   = o; o += (size_t)B * 9;
    const size_t VmO   = o; o += (size_t)B * 9;
    (void)ws_size; (void)n_in;

    float* outXm = out;                              // B*T*3
    float* outW  = out + (size_t)B * T * 3;          // B*9
    float* outS  = outW + (size_t)B * 9;             // B*3*T

    // 1) spatial mean (bandwidth-bound streaming pass over x, b128 loads)
    mean_kernel<<<B * 3 * T, 256, 0, stream>>>(x, outXm, ws + xmBT, B, T, (int)HW,
                                               1.0f / (float)HW);
    // 2) pentadiagonal LDL^T factorization (depends only on T)
    penta_factor<<<1, 1, 0, stream>>>(ws + facL1, ws + facL2, ws + facDD, T);
    // 3) detrend solves, one thread per (b,c) column
    {
        int BC = B * 3;
        penta_solve<<<(BC + 63) / 64, 64, 0, stream>>>(ws + xmBT, ws + facL1, ws + facL2,
                                                       ws + facDD, ws + detB, BC, T);
    }
    // 4) mean/std per channel
    stats_kernel<<<B * 3, 256, 0, stream>>>(ws + detB, ws + stats, T);
    // 5) zero the padded 16 x Tp staging buffers (Xpad,Ypad,Ppad contiguous)
    hipMemsetAsync(ws + Xpad, 0, (size_t)B * 16 * Tp * 3 * sizeof(float), stream);
    // 6) standardize into padded rows
    normalize_pad<<<B * 3, 256, 0, stream>>>(ws + detB, ws + stats, ws + Xpad, T, Tp);
    // 7) Rx = Xn Xn^T / T           (WMMA, one wave per batch)
    gram16_wmma<<<B, 32, 0, stream>>>(ws + Xpad, ws + RxO, Tp, 1.0f / (float)T);
    // 8) whitening from 3x3 eigh
    whiten3<<<(B + 63) / 64, 64, 0, stream>>>(ws + RxO, ws + WmO, ws + IWO, B);
    // 9) Y = W Xn
    ycompute<<<(B * T + 255) / 256, 256, 0, stream>>>(ws + Xpad, ws + WmO, ws + Ypad, B, T, Tp);
    // 10) R = Y Y^T / T             (WMMA)
    gram16_wmma<<<B, 32, 0, stream>>>(ws + Ypad, ws + RmO, Tp, 1.0f / (float)T);
    // 11) cumulant rows P
    pcompute<<<(B * T + 255) / 256, 256, 0, stream>>>(ws + Ypad, ws + Ppad, B, T, Tp);
    // 12) Qraw = P P^T / T          (WMMA, 9x9 in 16x16 tile)
    gram16_wmma<<<B, 32, 0, stream>>>(ws + Ppad, ws + QrO, Tp, 1.0f / (float)T);
    // 13) JADE: 9x9 eigh + 50 Jacobi sweeps + Wica = inv(IW V)
    jade_core<<<(B + 63) / 64, 64, 0, stream>>>(ws + QrO, ws + RmO, ws + IWO,
                                                ws + VmO, outW, B, T);
    // 14) S = V^T Y
    s_kernel<<<(B * T + 255) / 256, 256, 0, stream>>>(ws + Ypad, ws + VmO, outS, B, T, Tp);
}